// MPGNN_26929444946579
// MI455X (gfx1250) — compile-verified
//
#include <hip/hip_runtime.h>

#define NN   20000
#define EE   320000
#define INC  256
#define HID  256
#define OUTC 64
#define NG   64

typedef float v2f __attribute__((ext_vector_type(2)));
typedef float v8f __attribute__((ext_vector_type(8)));

// ---------------------------------------------------------------- utilities
__global__ __launch_bounds__(256) void mpgnn_fill0(float4* __restrict__ p, int n4) {
    int i = blockIdx.x * blockDim.x + threadIdx.x;
    if (i < n4) p[i] = make_float4(0.f, 0.f, 0.f, 0.f);
}

__global__ __launch_bounds__(256) void mpgnn_tanh4(float4* __restrict__ p, int n4) {
    int i = blockIdx.x * blockDim.x + threadIdx.x;
    if (i < n4) {
        float4 v = p[i];
        p[i] = make_float4(tanhf(v.x), tanhf(v.y), tanhf(v.z), tanhf(v.w));
    }
}

// --------------------------------------------------------------------------
// WMMA f32 GEMM, 16x64 strip per wave (4 accumulators, A reused across 4 WMMA)
// C[N,256] = A[N,256] @ W^T  (W is [256,256] row-major [out,in])
// tiles: tm in 0..1249 (M), tg in 0..3 (64-col group); 4 waves/block -> 1250 blocks
// LAYER=0: C = A@W^T            (x_proj)
// LAYER=1: C = relu(P + A@W^T)  (layer update; A already tanh'd)
// --------------------------------------------------------------------------
template <int LAYER>
__device__ __forceinline__ void mpgnn_gemm_body(
    const float* __restrict__ A, const float* __restrict__ W,
    const float* __restrict__ P, float* __restrict__ C) {
    int lane  = threadIdx.x & 31;
    int wave  = threadIdx.x >> 5;
    int tile  = blockIdx.x * 4 + wave;
    int tm    = tile >> 2;           // 0..1249
    int tg    = tile & 3;            // 0..3  (cols tg*64 .. tg*64+63)
    int half  = lane >> 4;           // 0 or 1
    int idx16 = lane & 15;

    // A operand: lane L(0-15): M=L, K={k,k+1}; lane L(16-31): M=L-16, K={k+2,k+3}
    const float* Arow  = A + (size_t)(tm * 16 + idx16) * HID + 2 * half;
    // B[k][n] = W[n][k]; column tile j adds 16 rows of W = 16*HID floats
    const float* Bbase = W + (size_t)(tg * 64 + idx16) * HID + 2 * half;

    v8f acc0 = {}, acc1 = {}, acc2 = {}, acc3 = {};
#pragma unroll 4
    for (int k = 0; k < HID; k += 4) {
        float2 af = *(const float2*)(Arow + k);
        v2f a; a[0] = af.x; a[1] = af.y;
        float2 b0f = *(const float2*)(Bbase + k);
        float2 b1f = *(const float2*)(Bbase + k + 16 * HID);
        float2 b2f = *(const float2*)(Bbase + k + 32 * HID);
        float2 b3f = *(const float2*)(Bbase + k + 48 * HID);
        v2f b0; b0[0] = b0f.x; b0[1] = b0f.y;
        v2f b1; b1[0] = b1f.x; b1[1] = b1f.y;
        v2f b2; b2[0] = b2f.x; b2[1] = b2f.y;
        v2f b3; b3[0] = b3f.x; b3[1] = b3f.y;
        acc0 = __builtin_amdgcn_wmma_f32_16x16x4_f32(false, a, false, b0, (short)0, acc0, false, false);
        acc1 = __builtin_amdgcn_wmma_f32_16x16x4_f32(false, a, false, b1, (short)0, acc1, false, false);
        acc2 = __builtin_amdgcn_wmma_f32_16x16x4_f32(false, a, false, b2, (short)0, acc2, false, false);
        acc3 = __builtin_amdgcn_wmma_f32_16x16x4_f32(false, a, false, b3, (short)0, acc3, false, false);
    }

    // D layout: VGPR i, lanes 0-15 -> M=i; lanes 16-31 -> M=i+8; N=lane&15
    int row0 = tm * 16 + half * 8;
    int col0 = tg * 64 + idx16;
    v8f accs[4] = {acc0, acc1, acc2, acc3};
#pragma unroll
    for (int j = 0; j < 4; ++j) {
#pragma unroll
        for (int i = 0; i < 8; ++i) {
            size_t o = (size_t)(row0 + i) * HID + col0 + j * 16;
            if (LAYER)
                C[o] = fmaxf(P[o] + accs[j][i], 0.f);
            else
                C[o] = accs[j][i];
        }
    }
}

__global__ __launch_bounds__(128) void mpgnn_gemm_xproj(
    const float* __restrict__ X, const float* __restrict__ W,
    float* __restrict__ P) {
    mpgnn_gemm_body<0>(X, W, nullptr, P);
}

__global__ __launch_bounds__(128) void mpgnn_gemm_layer(
    const float* __restrict__ AGG, const float* __restrict__ W,
    const float* __restrict__ P, float* __restrict__ H) {
    mpgnn_gemm_body<1>(AGG, W, P, H);
}

// -------------------------------------------------- edge gather + scatter-add
// agg[dst] += tanh(h[src]); one block (64 lanes = 2 waves) per edge, float4/lane
__global__ __launch_bounds__(64) void mpgnn_scatter(
    const float* __restrict__ H, const int* __restrict__ src,
    const int* __restrict__ dst, float* __restrict__ AGG) {
    int e = blockIdx.x;
    int s = src[e];          // uniform -> scalar load
    int d = dst[e];
    int c = threadIdx.x << 2;
    float4 v = *(const float4*)(H + (size_t)s * HID + c);
    float* o = AGG + (size_t)d * HID + c;
    unsafeAtomicAdd(o + 0, tanhf(v.x));
    unsafeAtomicAdd(o + 1, tanhf(v.y));
    unsafeAtomicAdd(o + 2, tanhf(v.z));
    unsafeAtomicAdd(o + 3, tanhf(v.w));
}

// ------------------------------------------------------------- mean pooling
__global__ __launch_bounds__(64) void mpgnn_pool(
    const float* __restrict__ H, const int* __restrict__ batch,
    float* __restrict__ PS, float* __restrict__ CNT) {
    int n = blockIdx.x;
    int g = batch[n];        // uniform
    int c = threadIdx.x << 2;
    float4 v = *(const float4*)(H + (size_t)n * HID + c);
    float* o = PS + (size_t)g * HID + c;
    unsafeAtomicAdd(o + 0, v.x);
    unsafeAtomicAdd(o + 1, v.y);
    unsafeAtomicAdd(o + 2, v.z);
    unsafeAtomicAdd(o + 3, v.w);
    if (threadIdx.x == 0) unsafeAtomicAdd(CNT + g, 1.0f);
}

// ---------------------------------------------------------------- prediction
// logits[g,o] = (pooled_sum[g]/max(cnt,1)) . Wpred[o] + b[o]
__global__ __launch_bounds__(256) void mpgnn_pred(
    const float* __restrict__ PS, const float* __restrict__ CNT,
    const float* __restrict__ Wp, const float* __restrict__ bp,
    float* __restrict__ LG) {
    int idx = blockIdx.x * blockDim.x + threadIdx.x;   // 4096
    int g = idx >> 6;
    int o = idx & 63;
    float inv = 1.0f / fmaxf(CNT[g], 1.0f);
    const float* pr = PS + (size_t)g * HID;
    const float* wr = Wp + (size_t)o * HID;
    float s = 0.f;
#pragma unroll 8
    for (int k = 0; k < HID; ++k) s += pr[k] * wr[k];
    LG[(size_t)g * OUTC + o] = s * inv + bp[o];
}

// ----------------------------------------------------- log_softmax (1 wave/row)
__global__ __launch_bounds__(32) void mpgnn_lsm(
    const float* __restrict__ LG, float* __restrict__ OUT) {
    int g = blockIdx.x;
    int l = threadIdx.x;                      // 0..31, 2 cols per lane
    float v0 = LG[(size_t)g * OUTC + l];
    float v1 = LG[(size_t)g * OUTC + 32 + l];
    float m = fmaxf(v0, v1);
#pragma unroll
    for (int off = 16; off; off >>= 1) m = fmaxf(m, __shfl_xor(m, off, 32));
    float s = __expf(v0 - m) + __expf(v1 - m);
#pragma unroll
    for (int off = 16; off; off >>= 1) s += __shfl_xor(s, off, 32);
    float lse = m + __logf(s);
    OUT[(size_t)g * OUTC + l]      = v0 - lse;
    OUT[(size_t)g * OUTC + 32 + l] = v1 - lse;
}

// ---------------------------------------------------------------------------
extern "C" void kernel_launch(void* const* d_in, const int* in_sizes, int n_in,
                              void* d_out, int out_size, void* d_ws, size_t ws_size,
                              hipStream_t stream) {
    const float* x     = (const float*)d_in[0];
    const int*   ei    = (const int*)d_in[1];   // [2,E]: row0=src, row1=dst
    const int*   batch = (const int*)d_in[2];
    const float* Wself = (const float*)d_in[3];
    const float* Wnbr  = (const float*)d_in[4];
    const float* Wpred = (const float*)d_in[5];
    const float* bpred = (const float*)d_in[6];
    float*       out   = (float*)d_out;

    const size_t NH = (size_t)NN * HID;          // 5,120,000 floats
    char* ws = (char*)d_ws;
    float* xproj  = (float*)ws; ws += NH * sizeof(float);
    float* h      = (float*)ws; ws += NH * sizeof(float);
    float* agg    = (float*)ws; ws += NH * sizeof(float);
    float* ps     = (float*)ws; ws += (size_t)NG * HID * sizeof(float);
    float* cnt    = (float*)ws; ws += 256;
    float* logits = (float*)ws;

    const int n4      = (int)(NH / 4);               // 1,280,000
    const int fillBlk = (n4 + 255) / 256;            // 5000
    const int gemmBlk = (NN / 16) * (HID / 64) / 4;  // 1250*4/4 = 1250

    // init accumulators (ws is poisoned by harness)
    mpgnn_fill0<<<fillBlk, 256, 0, stream>>>((float4*)h, n4);
    mpgnn_fill0<<<(NG * HID / 4 + 255) / 256, 256, 0, stream>>>((float4*)ps, NG * HID / 4);
    mpgnn_fill0<<<1, 32, 0, stream>>>((float4*)cnt, NG / 4);

    // x_proj = x @ W_self^T  (reused every layer)
    mpgnn_gemm_xproj<<<gemmBlk, 128, 0, stream>>>(x, Wself, xproj);

    for (int it = 0; it < 3; ++it) {
        mpgnn_fill0<<<fillBlk, 256, 0, stream>>>((float4*)agg, n4);
        mpgnn_scatter<<<EE, 64, 0, stream>>>(h, ei, ei + EE, agg);
        mpgnn_tanh4<<<fillBlk, 256, 0, stream>>>((float4*)agg, n4);  // out = tanh(agg)
        mpgnn_gemm_layer<<<gemmBlk, 128, 0, stream>>>(agg, Wnbr, xproj, h);
    }

    mpgnn_pool<<<NN, 64, 0, stream>>>(h, batch, ps, cnt);
    mpgnn_pred<<<(NG * OUTC) / 256, 256, 0, stream>>>(ps, cnt, Wpred, bpred, logits);
    mpgnn_lsm<<<NG, 32, 0, stream>>>(logits, out);
}